// RoleEmbeddingManager_70025146794162
// MI455X (gfx1250) — compile-verified
//
#include <hip/hip_runtime.h>
#include <stdint.h>
#include <stddef.h>

#ifndef __has_builtin
#define __has_builtin(x) 0
#endif

// gfx1250 async global->LDS path (ASYNCcnt-tracked), with safe fallback.
#if defined(__AMDGCN__) && __has_builtin(__builtin_amdgcn_global_load_async_to_lds_b128)
#define USE_ASYNC_LDS 1
#else
#define USE_ASYNC_LDS 0
#endif

typedef __attribute__((ext_vector_type(4))) float v4f;
// Exact pointee type of the async-LDS builtin per hipcc's diagnostic:
// 'int __attribute__((vector_size(4*sizeof(int))))'
typedef int b128_t __attribute__((vector_size(16)));

#define AS_GLOBAL __attribute__((address_space(1)))
#define AS_LDS    __attribute__((address_space(3)))

// Problem constants (match reference): B=4096, N=128, T=64, R=16, D=128.
constexpr int kN      = 128;
constexpr int kR      = 16;
constexpr int kD      = 128;
constexpr int kRowV4  = kD / 4;          // 32 float4 per embedding row (512 B)
constexpr int kTabV4  = kR * kRowV4;     // 512 float4 = 8 KiB sub-table
constexpr int kBlock  = 256;             // 8 wave32s per block
constexpr int kOutV4  = kN * kRowV4;     // 4096 float4 of output per batch row
constexpr int kIters  = kOutV4 / kBlock; // 16

__global__ __launch_bounds__(kBlock) void role_gather_kernel(
    const int*   __restrict__ template_id,  // [B]
    const int*   __restrict__ atom_role,    // [B, N]
    const float* __restrict__ tables,       // [T, R, D]
    float*       __restrict__ out)          // [B, N, D]
{
    __shared__ v4f s_table[kTabV4];  // 8 KiB: tables[t] staged in LDS
    __shared__ int s_role[kN];       // 512 B: roles for this batch row

    const int b   = blockIdx.x;
    const int tid = threadIdx.x;
    const int t   = template_id[b];

    const v4f* __restrict__ src =
        (const v4f*)(tables + (size_t)t * (kR * kD));

#if USE_ASYNC_LDS
    // Async DMA of the 8 KiB sub-table straight into LDS (no VGPR round trip).
    // 256 lanes x 2 x b128 = 8 KiB.
    __builtin_amdgcn_global_load_async_to_lds_b128(
        (AS_GLOBAL b128_t*)(src + tid),
        (AS_LDS    b128_t*)(s_table + tid),
        0, 0);
    __builtin_amdgcn_global_load_async_to_lds_b128(
        (AS_GLOBAL b128_t*)(src + tid + kBlock),
        (AS_LDS    b128_t*)(s_table + tid + kBlock),
        0, 0);
#else
    s_table[tid]          = src[tid];
    s_table[tid + kBlock] = src[tid + kBlock];
#endif

    // Stage the 128 role indices while the table copy is in flight.
    if (tid < kN) s_role[tid] = atom_role[(size_t)b * kN + tid];

#if USE_ASYNC_LDS
#if __has_builtin(__builtin_amdgcn_s_wait_asynccnt)
    __builtin_amdgcn_s_wait_asynccnt(0);
#else
    asm volatile("s_wait_asynccnt 0" ::: "memory");
#endif
#endif
    __syncthreads();

    v4f* __restrict__ out4 = (v4f*)(out + (size_t)b * (kN * kD));

    // Each wave32 emits whole 512-B rows: for a given k, all 32 lanes share
    // n (wave-uniform role -> LDS broadcast read of s_role), read 32
    // consecutive b128 words from the LDS row (bank-conflict-free), and issue
    // one fully-coalesced 512-B non-temporal global_store_b128 per wave.
#pragma unroll
    for (int k = 0; k < kIters; ++k) {
        const int f    = tid + k * kBlock;   // flat float4 index, 0..4095
        const int n    = f >> 5;             // atom index (wave-uniform)
        const int j    = f & 31;             // float4 within row (== lane)
        const int role = s_role[n];
        v4f val = s_table[role * kRowV4 + j];
        __builtin_nontemporal_store(val, out4 + f);  // output >> L2, stream it
    }
}

extern "C" void kernel_launch(void* const* d_in, const int* in_sizes, int n_in,
                              void* d_out, int out_size, void* d_ws, size_t ws_size,
                              hipStream_t stream) {
    const int*   template_id = (const int*)d_in[0];   // [B] int32
    const int*   atom_role   = (const int*)d_in[1];   // [B,N] int32
    const float* tables      = (const float*)d_in[2]; // [T,R,D] fp32
    float*       out         = (float*)d_out;         // [B,N,D] fp32

    const int B = in_sizes[0];  // 4096 blocks, one per batch row
    role_gather_kernel<<<B, kBlock, 0, stream>>>(template_id, atom_role, tables, out);
}